// AttnGCN_15375982919974
// MI455X (gfx1250) — compile-verified
//
#include <hip/hip_runtime.h>
#include <hip/hip_bf16.h>
#include <math.h>

// ---------------- problem constants ----------------
#define Dc 30
#define Hh 4
#define Cc 30
#define HC 120
#define NEG_SLOPE 0.2f

typedef float v2f __attribute__((ext_vector_type(2)));
typedef float v8f __attribute__((ext_vector_type(8)));

__device__ __forceinline__ v8f wmma4(v2f a, v2f b, v8f c) {
    // V_WMMA_F32_16X16X4_F32 : full fp32 matrix op, 16x16 D tile, K=4
    return __builtin_amdgcn_wmma_f32_16x16x4_f32(
        false, a, false, b, (short)0, c, false, false);
}

__device__ __forceinline__ void atomicMaxF(float* addr, float val) {
    val += 0.0f;  // canonicalize -0 -> +0
    if (val >= 0.0f)
        atomicMax((int*)addr, __float_as_int(val));
    else
        atomicMin((unsigned int*)addr, __float_as_uint(val));
}

// ---------------- 1: init workspace ----------------
__global__ void k_init(float* mws, float* den, float* agg, float* stats, int N) {
    int tid = blockIdx.x * blockDim.x + threadIdx.x;
    int stride = gridDim.x * blockDim.x;
    int n4 = N * Hh, n30 = N * Cc;
    for (int i = tid; i < n30; i += stride) agg[i] = 0.0f;
    for (int i = tid; i < n4; i += stride) {
        mws[i] = __int_as_float(0xff800000);  // -inf
        den[i] = 0.0f;
    }
    if (tid < 256) stats[tid] = 0.0f;
}

// ---------------- 2: column mean/var stats + edge-weight sum ----------------
__global__ void k_stats(const float* __restrict__ hin, const float* __restrict__ ew,
                        float* stats, int N, int E) {
    __shared__ float ssum[Dc], ssq[Dc], sew;
    if (threadIdx.x < Dc) { ssum[threadIdx.x] = 0.f; ssq[threadIdx.x] = 0.f; }
    if (threadIdx.x == 0) sew = 0.f;
    __syncthreads();
    float ls[Dc], lq[Dc];
#pragma unroll
    for (int c = 0; c < Dc; ++c) { ls[c] = 0.f; lq[c] = 0.f; }
    int tid = blockIdx.x * blockDim.x + threadIdx.x;
    int stride = gridDim.x * blockDim.x;
    for (int r = tid; r < N; r += stride) {
        const float* row = hin + (size_t)r * Dc;
#pragma unroll
        for (int c = 0; c < Dc; ++c) { float v = row[c]; ls[c] += v; lq[c] += v * v; }
    }
    float le = 0.f;
    for (int e = tid; e < E; e += stride) le += ew[e];
#pragma unroll
    for (int c = 0; c < Dc; ++c) { atomicAdd(&ssum[c], ls[c]); atomicAdd(&ssq[c], lq[c]); }
    atomicAdd(&sew, le);
    __syncthreads();
    if (threadIdx.x < Dc) {
        atomicAdd(&stats[threadIdx.x], ssum[threadIdx.x]);
        atomicAdd(&stats[Dc + threadIdx.x], ssq[threadIdx.x]);
    }
    if (threadIdx.x == 0) atomicAdd(&stats[60], sew);
}

// ---------------- 3: fold BN into W' (32x128 padded) + edge-att scalars ----------
__global__ void k_fold(const float* __restrict__ W, const float* __restrict__ gamma,
                       const float* __restrict__ beta, const float* __restrict__ Wedge,
                       const float* __restrict__ att_edge, float* stats,
                       float* Wp, float* bp, int N, int E) {
    __shared__ float sscale[Dc], sshift[Dc];
    int t = threadIdx.x;
    if (t < Dc) {
        float mu = stats[t] / (float)N;
        float var = stats[Dc + t] / (float)N - mu * mu;
        float inv = rsqrtf(var + 1e-5f);
        float sc = inv * gamma[t];
        sscale[t] = sc;
        sshift[t] = beta[t] - mu * sc;
    }
    if (t == 0) stats[61] = stats[60] / (float)E;  // mean edge weight
    if (t < Hh) {                                   // a_edge scalar per head
        float s = 0.f;
        for (int c = 0; c < Cc; ++c) s += Wedge[t * Cc + c] * att_edge[t * Cc + c];
        stats[62 + t] = s;
    }
    __syncthreads();
    if (t < 128) {
        float b = 0.f;
        for (int k = 0; k < 32; ++k) {
            float w = (k < Dc && t < HC) ? sscale[k] * W[k * HC + t] : 0.f;
            Wp[k * 128 + t] = w;
            if (k < Dc && t < HC) b += sshift[k] * W[k * HC + t];
        }
        bp[t] = (t < HC) ? b : 0.f;
    }
}

// ---------------- 4: WMMA GEMM  x = h @ W' + b'  (N x 120) --------------------
// B staged in WMMA fragment order: one float2 per (frag, lane) so each B operand
// is a single conflict-free ds_load_b64 (32 consecutive float2 = 256B = 64 banks).
__global__ __launch_bounds__(256) void k_gemm_x(const float* __restrict__ hin,
                                                const float* __restrict__ Wp,
                                                const float* __restrict__ bp,
                                                float* __restrict__ xws, int N) {
    __shared__ v2f sWf[2048];  // frag f = kk*8 + t ; sWf[f*32 + lane]
    for (int j = threadIdx.x; j < 2048; j += 256) {
        int f = j >> 5, lane = j & 31;
        int kk = f >> 3, tt = f & 7;
        int half = lane >> 4, l = lane & 15;
        int k = kk * 4 + 2 * half, n = tt * 16 + l;
        v2f w;
        w.x = Wp[k * 128 + n];
        w.y = Wp[(k + 1) * 128 + n];
        sWf[j] = w;
    }
    __syncthreads();
    int wv = threadIdx.x >> 5, lane = threadIdx.x & 31;
    int half = lane >> 4, l = lane & 15;
    int m0 = blockIdx.x * 128 + wv * 16;
    int row = m0 + l;
    // Clamp: OOB rows read valid (finite) data; their accumulator rows are never
    // stored. K-pad cols 30/31 multiply B rows that are exactly zero.
    const float* rowp = hin + (size_t)(row < N ? row : N - 1) * Dc;
    v8f acc[8] = {};
#pragma unroll
    for (int kk = 0; kk < 8; ++kk) {
        int ko = kk * 4 + 2 * half;
        if (ko > 28) ko = 28;                    // only kk==7/half==1; B rows are 0 there
        v2f a = *(const v2f*)(rowp + ko);        // unconditional global_load_b64
#pragma unroll
        for (int t = 0; t < 8; ++t) {
            v2f b = sWf[(kk * 8 + t) * 32 + lane];
            acc[t] = wmma4(a, b, acc[t]);
        }
    }
    if (m0 + 16 <= N) {  // uniform fast path: no row guards
#pragma unroll
        for (int t = 0; t < 8; ++t) {
            int col = t * 16 + l;
            float bb = bp[col];  // bp padded to 128
#pragma unroll
            for (int r = 0; r < 8; ++r) {
                int ro = m0 + r + 8 * half;  // D layout: vgpr r -> M=r (+8 hi half)
                if (t < 7 || l < 8) xws[(size_t)ro * HC + col] = acc[t][r] + bb;
            }
        }
    } else if (m0 < N) {
#pragma unroll
        for (int t = 0; t < 8; ++t) {
            int col = t * 16 + l;
            float bb = bp[col];
#pragma unroll
            for (int r = 0; r < 8; ++r) {
                int ro = m0 + r + 8 * half;
                if ((t < 7 || l < 8) && ro < N)
                    xws[(size_t)ro * HC + col] = acc[t][r] + bb;
            }
        }
    }
}

// ---------------- 5: attention dot products per (node, head) -----------------
__global__ void k_attdots(const float* __restrict__ xws, const float* __restrict__ att_src,
                          const float* __restrict__ att_dst, float* asrc, float* adst, int N) {
    int t = blockIdx.x * blockDim.x + threadIdx.x;
    if (t >= N * Hh) return;
    int n = t >> 2, hh = t & 3;
    const float* xr = xws + (size_t)n * HC + hh * Cc;
    float s1 = 0.f, s2 = 0.f;
#pragma unroll
    for (int c = 0; c < Cc; ++c) {
        float v = xr[c];
        s1 += v * att_src[hh * Cc + c];
        s2 += v * att_dst[hh * Cc + c];
    }
    asrc[t] = s1;
    adst[t] = s2;
}

// ---------------- edge helpers ----------------
__device__ __forceinline__ void edge_sdw(const int* ei, const float* ew, const float* stats,
                                         int e, int E, int& s, int& d, float& w) {
    if (e < E) { s = ei[e]; d = ei[E + e]; w = ew[e]; }
    else       { s = e - E; d = s; w = stats[61]; }   // self loop, mean weight
}

// ---------------- 6: segment max of leaky-relu logits -------------------------
__global__ void k_edge_max(const int* __restrict__ ei, const float* __restrict__ ew,
                           const float* __restrict__ asrc, const float* __restrict__ adst,
                           const float* __restrict__ stats, float* mws, int N, int E) {
    int t = blockIdx.x * blockDim.x + threadIdx.x;
    if (t >= E + N) return;
    int s, d; float w;
    edge_sdw(ei, ew, stats, t, E, s, d, w);
#pragma unroll
    for (int h = 0; h < Hh; ++h) {
        float a = asrc[s * Hh + h] + adst[d * Hh + h] + w * stats[62 + h];
        a = (a > 0.f) ? a : NEG_SLOPE * a;
        atomicMaxF(&mws[d * Hh + h], a);
    }
}

// ---------------- 7: segment sum of exp ---------------------------------------
__global__ void k_edge_den(const int* __restrict__ ei, const float* __restrict__ ew,
                           const float* __restrict__ asrc, const float* __restrict__ adst,
                           const float* __restrict__ stats, const float* __restrict__ mws,
                           float* den, int N, int E) {
    int t = blockIdx.x * blockDim.x + threadIdx.x;
    if (t >= E + N) return;
    int s, d; float w;
    edge_sdw(ei, ew, stats, t, E, s, d, w);
#pragma unroll
    for (int h = 0; h < Hh; ++h) {
        float a = asrc[s * Hh + h] + adst[d * Hh + h] + w * stats[62 + h];
        a = (a > 0.f) ? a : NEG_SLOPE * a;
        atomicAdd(&den[d * Hh + h], expf(a - mws[d * Hh + h]));
    }
}

// ---------------- 8: weighted aggregation (one wave32 per edge) ---------------
__global__ void k_edge_agg(const int* __restrict__ ei, const float* __restrict__ ew,
                           const float* __restrict__ asrc, const float* __restrict__ adst,
                           const float* __restrict__ stats, const float* __restrict__ mws,
                           const float* __restrict__ den, const float* __restrict__ xws,
                           float* agg, int N, int E) {
    int gt = blockIdx.x * blockDim.x + threadIdx.x;
    int e = gt >> 5, c = gt & 31;
    if (e >= E + N) return;
    int s, d; float w;
    edge_sdw(ei, ew, stats, e, E, s, d, w);
    // start pulling the gathered x row (L2 resident) while alpha math runs
    __builtin_prefetch(&xws[(size_t)s * HC + c], 0, 1);
    float val = 0.f;
#pragma unroll
    for (int h = 0; h < Hh; ++h) {
        float a = asrc[s * Hh + h] + adst[d * Hh + h] + w * stats[62 + h];
        a = (a > 0.f) ? a : NEG_SLOPE * a;
        float al = expf(a - mws[d * Hh + h]) / (den[d * Hh + h] + 1e-16f);
        if (c < Cc) val += xws[(size_t)s * HC + h * Cc + c] * al;
    }
    if (c < Cc) atomicAdd(&agg[(size_t)d * Cc + c], 0.25f * val);  // 0.25 = head mean
}

// ---------------- 9: WMMA MLP chain: relu(agg+bias) -> fc1 -> fc2 -> fc3 ------
__global__ __launch_bounds__(256) void k_mlp(const float* __restrict__ agg,
                                             const float* __restrict__ bias,
                                             const float* __restrict__ fc1w, const float* __restrict__ fc1b,
                                             const float* __restrict__ fc2w, const float* __restrict__ fc2b,
                                             const float* __restrict__ fc3w, const float* __restrict__ fc3b,
                                             float* __restrict__ out, int N) {
    __shared__ v2f Bf[3 * 16 * 32];     // fragment order: (lyr*16 + kk*2 + t)*32 + lane
    __shared__ float Ast[8 * 16 * 34];  // per-wave activation tile, even stride -> b64 A reads
    for (int j = threadIdx.x; j < 1536; j += 256) {
        int lyr = j >> 9, rem = j & 511;
        int f = rem >> 5, lane = rem & 31;
        int kk = f >> 1, tt = f & 1;
        int half = lane >> 4, l = lane & 15;
        int k = kk * 4 + 2 * half, n = tt * 16 + l;
        const float* src = (lyr == 0) ? fc1w : (lyr == 1) ? fc2w : fc3w;
        v2f w;
        w.x = (k < Cc && n < Cc) ? src[n * Cc + k] : 0.f;
        w.y = (k + 1 < Cc && n < Cc) ? src[n * Cc + k + 1] : 0.f;
        Bf[j] = w;
    }
    int wv = threadIdx.x >> 5, lane = threadIdx.x & 31;
    int half = lane >> 4, l = lane & 15;
    int m0 = blockIdx.x * 128 + wv * 16;
    float* As = &Ast[wv * 544];
    for (int i = lane; i < 512; i += 32) {
        int r = i >> 5, cc = i & 31;
        int row = m0 + r;
        int ri = row < N ? row : N - 1;      // clamp: OOB rows never stored
        int ci = cc < Cc ? cc : 0;           // no speculative OOB loads
        float v = fmaxf(agg[(size_t)ri * Cc + ci] + bias[ci], 0.f);
        As[r * 34 + cc] = (cc < Cc) ? v : 0.f;  // K-pad cols 30/31 = 0
    }
    __syncthreads();
    for (int lyr = 0; lyr < 3; ++lyr) {
        const v2f* Bl = &Bf[lyr * 512];
        const float* fb = (lyr == 0) ? fc1b : (lyr == 1) ? fc2b : fc3b;
        v8f acc[2] = {};
#pragma unroll
        for (int kk = 0; kk < 8; ++kk) {
            v2f a = *(const v2f*)&As[l * 34 + kk * 4 + 2 * half];  // ds_load_b64
#pragma unroll
            for (int t = 0; t < 2; ++t) {
                v2f b = Bl[(kk * 2 + t) * 32 + lane];
                acc[t] = wmma4(a, b, acc[t]);
            }
        }
        if (lyr < 2) {
#pragma unroll
            for (int t = 0; t < 2; ++t) {
                int col = t * 16 + l;
                int ci = col < Cc ? col : 0;
                float bb = fb[ci];
                bb = (col < Cc) ? bb : 0.f;  // keep pad cols exactly 0
#pragma unroll
                for (int r = 0; r < 8; ++r)
                    As[(r + 8 * half) * 34 + col] = fmaxf(acc[t][r] + bb, 0.f);
            }
            // in-wave LDS RAW: make write-back visible before next layer's reads
            asm volatile("s_wait_dscnt 0" ::: "memory");
        } else {
            if (m0 + 16 <= N) {  // uniform fast path
#pragma unroll
                for (int t = 0; t < 2; ++t) {
                    int col = t * 16 + l;
#pragma unroll
                    for (int r = 0; r < 8; ++r) {
                        int ro = m0 + r + 8 * half;
                        if (t == 0 || l < 14)
                            out[(size_t)ro * Cc + col] = acc[t][r] + fb[col];
                    }
                }
            } else if (m0 < N) {
#pragma unroll
                for (int t = 0; t < 2; ++t) {
                    int col = t * 16 + l;
#pragma unroll
                    for (int r = 0; r < 8; ++r) {
                        int ro = m0 + r + 8 * half;
                        if ((t == 0 || l < 14) && ro < N)
                            out[(size_t)ro * Cc + col] = acc[t][r] + fb[col];
                    }
                }
            }
        }
    }
}

// ---------------- launch ----------------
extern "C" void kernel_launch(void* const* d_in, const int* in_sizes, int n_in,
                              void* d_out, int out_size, void* d_ws, size_t ws_size,
                              hipStream_t stream) {
    const float* hin      = (const float*)d_in[0];
    const int*   ei       = (const int*)  d_in[1];
    const float* ew       = (const float*)d_in[2];
    const float* gamma    = (const float*)d_in[3];
    const float* beta     = (const float*)d_in[4];
    const float* W        = (const float*)d_in[5];
    const float* att_src  = (const float*)d_in[6];
    const float* att_dst  = (const float*)d_in[7];
    const float* att_edge = (const float*)d_in[8];
    const float* Wedge    = (const float*)d_in[9];
    const float* bias     = (const float*)d_in[10];
    const float* fc1w     = (const float*)d_in[11];
    const float* fc1b     = (const float*)d_in[12];
    const float* fc2w     = (const float*)d_in[13];
    const float* fc2b     = (const float*)d_in[14];
    const float* fc3w     = (const float*)d_in[15];
    const float* fc3b     = (const float*)d_in[16];
    float* out = (float*)d_out;

    const int N = in_sizes[0] / Dc;
    const int E = in_sizes[2];
    const int E2 = E + N;

    // workspace layout (floats): x fits in L2 (48MB) -> edge gathers are L2-bound
    float* ws = (float*)d_ws;
    size_t o = 0;
    float* XWS  = ws + o; o += (size_t)N * HC;   // x = hn @ W  [N,120]
    float* ASRC = ws + o; o += (size_t)N * Hh;
    float* ADST = ws + o; o += (size_t)N * Hh;
    float* MWS  = ws + o; o += (size_t)N * Hh;   // segment max
    float* DEN  = ws + o; o += (size_t)N * Hh;   // segment sum
    float* AGG  = ws + o; o += (size_t)N * Cc;   // head-mean aggregate
    float* WP   = ws + o; o += 32 * 128;         // BN-folded, zero-padded W'
    float* BP   = ws + o; o += 128;              // folded bias b'
    float* STATS= ws + o; o += 256;              // sums, mean_ew, per-head edge scalars

    dim3 blk(256);
    k_init<<<dim3(1024), blk, 0, stream>>>(MWS, DEN, AGG, STATS, N);
    k_stats<<<dim3(256), blk, 0, stream>>>(hin, ew, STATS, N, E);
    k_fold<<<dim3(1), dim3(128), 0, stream>>>(W, gamma, beta, Wedge, att_edge, STATS, WP, BP, N, E);

    int gemm_blocks = (N + 127) / 128;
    k_gemm_x<<<dim3(gemm_blocks), blk, 0, stream>>>(hin, WP, BP, XWS, N);
    k_attdots<<<dim3((N * Hh + 255) / 256), blk, 0, stream>>>(XWS, att_src, att_dst, ASRC, ADST, N);

    int eblocks = (E2 + 255) / 256;
    k_edge_max<<<dim3(eblocks), blk, 0, stream>>>(ei, ew, ASRC, ADST, STATS, MWS, N, E);
    k_edge_den<<<dim3(eblocks), blk, 0, stream>>>(ei, ew, ASRC, ADST, STATS, MWS, DEN, N, E);
    int ablocks = (int)(((size_t)E2 * 32 + 255) / 256);
    k_edge_agg<<<dim3(ablocks), blk, 0, stream>>>(ei, ew, ASRC, ADST, STATS, MWS, DEN, XWS, AGG, N, E);

    k_mlp<<<dim3(gemm_blocks), blk, 0, stream>>>(AGG, bias, fc1w, fc1b, fc2w, fc2b,
                                                 fc3w, fc3b, out, N);
}